// RRNNCell_26456998543616
// MI455X (gfx1250) — compile-verified
//
#include <hip/hip_runtime.h>
#include <stdint.h>

#ifndef __has_builtin
#define __has_builtin(x) 0
#endif

// ---- CDNA5 async global->LDS path (guarded so compile never hard-fails) ----
#if defined(__HIP_DEVICE_COMPILE__) && defined(__gfx1250__) && \
    __has_builtin(__builtin_amdgcn_global_load_async_to_lds_b128) && \
    __has_builtin(__builtin_amdgcn_s_wait_asynccnt)
#define HAS_ASYNC 1
#else
#define HAS_ASYNC 0
#endif

#define L_DIM 2048
#define B_DIM 32
#define D_DIM 256
#define C_DIM (B_DIM * D_DIM)   // 8192 channels
#define NC 32                   // chunks along L
#define CL (L_DIM / NC)         // 64 timesteps per chunk
#define TS 8                    // timesteps per async stage
#define NSTAGE (CL / TS)        // 8 stages per chunk

#if HAS_ASYNC
typedef int vint4 __attribute__((ext_vector_type(4)));
typedef __attribute__((address_space(1))) vint4 gvint4;
typedef __attribute__((address_space(3))) vint4 lvint4;
#endif

__device__ __forceinline__ void async_copy16(const float4* g, float4* l) {
#if HAS_ASYNC
  // 16B per lane, DMA into LDS; tracked by ASYNCcnt.
  __builtin_amdgcn_global_load_async_to_lds_b128((gvint4*)g, (lvint4*)l, 0, 0);
#else
  *l = *g;
#endif
}

template <int N>
__device__ __forceinline__ void async_wait() {
#if HAS_ASYNC
  __builtin_amdgcn_s_wait_asynccnt(N);
#endif
}

__device__ __forceinline__ void ds_reads_drain() {
#if HAS_ASYNC
#if __has_builtin(__builtin_amdgcn_s_wait_dscnt)
  __builtin_amdgcn_s_wait_dscnt(0);
#else
  asm volatile("s_wait_dscnt 0" ::: "memory");
#endif
#endif
}

// ---------------- Phase 1: per-chunk affine transfer functions -------------
// For a chunk, with start state (s1,s2):
//   c1_end = A*s1 + b1 ;  c2_end = Q*s1 + P*s2 + b2
__global__ __launch_bounds__(D_DIM) void p1_transfer(
    const float* __restrict__ u, const float* __restrict__ eps,
    float* __restrict__ ws) {
  __shared__ float4 buf[2][TS][D_DIM];  // 64 KB
  const int k = blockIdx.x / B_DIM;
  const int b = blockIdx.x - k * B_DIM;
  const int d = threadIdx.x;
  const float e = eps[d];

  // u[t][b][d] as float4 at linear index (t*B + b)*D + d ; t-stride = C_DIM
  const float4* g = (const float4*)u + ((size_t)(k * CL) * B_DIM + b) * D_DIM + d;

#pragma unroll
  for (int ts = 0; ts < TS; ++ts)
    async_copy16(g + (size_t)ts * C_DIM, &buf[0][ts][d]);

  float A = 1.f, P = 1.f, Q = 0.f, b1 = 0.f, b2 = 0.f;
  for (int s = 0; s < NSTAGE; ++s) {
    const int cur = s & 1;
    if (s + 1 < NSTAGE) {
      ds_reads_drain();  // buffer we refill was read by ds_loads in the previous stage
      const float4* gn = g + (size_t)(s + 1) * TS * C_DIM;
#pragma unroll
      for (int ts = 0; ts < TS; ++ts)
        async_copy16(gn + (size_t)ts * C_DIM, &buf[cur ^ 1][ts][d]);
      async_wait<TS>();  // oldest TS (current stage) complete; in-order retire
    } else {
      async_wait<0>();
    }
#pragma unroll
    for (int ts = 0; ts < TS; ++ts) {
      const float4 v = buf[cur][ts][d];  // (u1,u2,f1,f2)
      const float Qn  = fmaf(v.y, A, v.w * Q);
      A = v.z * A;
      P = v.w * P;
      const float b2n = fmaf(v.w, b2, v.y * (e + b1));
      b1 = fmaf(v.z, b1, v.x);
      Q = Qn; b2 = b2n;
    }
  }
  const size_t NCC = (size_t)NC * C_DIM;
  const size_t idx = (size_t)k * C_DIM + (size_t)b * D_DIM + d;
  ws[0 * NCC + idx] = A;
  ws[1 * NCC + idx] = P;
  ws[2 * NCC + idx] = Q;
  ws[3 * NCC + idx] = b1;
  ws[4 * NCC + idx] = b2;
}

// ---------------- Phase 2: fold chunk transfers -> chunk-start states ------
__global__ __launch_bounds__(256) void p2_fold(
    const float* __restrict__ c1i, const float* __restrict__ c2i,
    float* __restrict__ ws, float* __restrict__ c1f, float* __restrict__ c2f) {
  const int ch = blockIdx.x * 256 + threadIdx.x;  // b*D + d
  const size_t NCC = (size_t)NC * C_DIM;
  float s1 = c1i[ch];
  float s2 = c2i[ch];
  for (int k = 0; k < NC; ++k) {
    const size_t idx = (size_t)k * C_DIM + ch;
    ws[5 * NCC + idx] = s1;
    ws[6 * NCC + idx] = s2;
    const float A  = ws[0 * NCC + idx];
    const float P  = ws[1 * NCC + idx];
    const float Q  = ws[2 * NCC + idx];
    const float b1 = ws[3 * NCC + idx];
    const float b2 = ws[4 * NCC + idx];
    const float s1n = fmaf(A, s1, b1);
    const float s2n = fmaf(Q, s1, fmaf(P, s2, b2));
    s1 = s1n; s2 = s2n;
  }
  c1f[ch] = s1;
  c2f[ch] = s2;
}

// ---------------- Phase 3: re-scan chunks from exact start, stream out -----
__global__ __launch_bounds__(D_DIM) void p3_scan(
    const float* __restrict__ u, const float* __restrict__ eps,
    const float* __restrict__ ws,
    float* __restrict__ c1s, float* __restrict__ c2s) {
  __shared__ float4 buf[2][TS][D_DIM];  // 64 KB
  const int k = blockIdx.x / B_DIM;
  const int b = blockIdx.x - k * B_DIM;
  const int d = threadIdx.x;
  const float e = eps[d];

  const float4* g = (const float4*)u + ((size_t)(k * CL) * B_DIM + b) * D_DIM + d;

#pragma unroll
  for (int ts = 0; ts < TS; ++ts)
    async_copy16(g + (size_t)ts * C_DIM, &buf[0][ts][d]);

  const size_t NCC = (size_t)NC * C_DIM;
  const size_t sidx = (size_t)k * C_DIM + (size_t)b * D_DIM + d;
  float c1 = ws[5 * NCC + sidx];
  float c2 = ws[6 * NCC + sidx];

  float* o1 = c1s + (size_t)(k * CL) * C_DIM + (size_t)b * D_DIM + d;
  float* o2 = c2s + (size_t)(k * CL) * C_DIM + (size_t)b * D_DIM + d;

  for (int s = 0; s < NSTAGE; ++s) {
    const int cur = s & 1;
    if (s + 1 < NSTAGE) {
      ds_reads_drain();
      const float4* gn = g + (size_t)(s + 1) * TS * C_DIM;
#pragma unroll
      for (int ts = 0; ts < TS; ++ts)
        async_copy16(gn + (size_t)ts * C_DIM, &buf[cur ^ 1][ts][d]);
      async_wait<TS>();
    } else {
      async_wait<0>();
    }
#pragma unroll
    for (int ts = 0; ts < TS; ++ts) {
      const float4 v = buf[cur][ts][d];  // (u1,u2,f1,f2)
      const float c1n = fmaf(c1, v.z, v.x);
      const float c2n = fmaf(c2, v.w, v.y * (e + c1));
      c1 = c1n; c2 = c2n;
      const size_t off = (size_t)(s * TS + ts) * C_DIM;
      __builtin_nontemporal_store(c1, o1 + off);  // write-once stream
      __builtin_nontemporal_store(c2, o2 + off);
    }
  }
}

extern "C" void kernel_launch(void* const* d_in, const int* in_sizes, int n_in,
                              void* d_out, int out_size, void* d_ws, size_t ws_size,
                              hipStream_t stream) {
  (void)in_sizes; (void)n_in; (void)out_size; (void)ws_size;
  const float* u   = (const float*)d_in[0];  // [L,B,1,D,4]
  const float* c1i = (const float*)d_in[1];  // [B,D]
  const float* c2i = (const float*)d_in[2];  // [B,D]
  const float* eps = (const float*)d_in[3];  // [1,D]

  float* out = (float*)d_out;
  float* ws  = (float*)d_ws;  // needs 7 * NC * C_DIM floats = ~7.4 MB

  const size_t LBD = (size_t)L_DIM * C_DIM;
  float* c1s = out;                       // [L,B,1,D]
  float* c2s = out + LBD;                 // [L,B,1,D]
  float* c1f = out + 2 * LBD;             // [B,D]
  float* c2f = out + 2 * LBD + C_DIM;     // [B,D]

  p1_transfer<<<dim3(NC * B_DIM), dim3(D_DIM), 0, stream>>>(u, eps, ws);
  p2_fold<<<dim3(C_DIM / 256), dim3(256), 0, stream>>>(c1i, c2i, ws, c1f, c2f);
  p3_scan<<<dim3(NC * B_DIM), dim3(D_DIM), 0, stream>>>(u, eps, ws, c1s, c2s);
}